// OEPEmbedding_49065706390109
// MI455X (gfx1250) — compile-verified
//
#include <hip/hip_runtime.h>
#include <cstdint>

// TDM descriptor SGPR-group types (clang-23 / amdgpu-toolchain 6-arg builtin form:
//   (uint32x4 g0, int32x8 g1, int32x4 g2, int32x4 g3, int32x8 g4, i32 cpol) )
typedef __attribute__((ext_vector_type(4))) unsigned int v4u;
typedef __attribute__((ext_vector_type(8))) int          v8i;
typedef __attribute__((ext_vector_type(4))) int          v4i;

#define N_GRAMS   3
#define EMBED_DIM 2048
#define VOCAB_N   128000
#define ROWS_PER_BLOCK 4   // 4 rows x 8KB = 32KB LDS tile; <=8 idx per gather D#

// ---- TDM gather load: pull `nrows` indexed 2048-float rows of `wbase` into LDS ----
__device__ __forceinline__ void tdm_gather_load(const float* wbase, uint32_t lds_off,
                                                int nrows, int i0, int i1, int i2, int i3) {
    uint64_t ga = (uint64_t)(uintptr_t)wbase;
    v4u g0;
    g0.x = 1u | (1u << 30) | (1u << 31);              // count=1, 32-bit indices, gather_mode
    g0.y = lds_off;                                    // lds_addr
    g0.z = (uint32_t)(ga & 0xFFFFFFFFu);               // global_addr[31:0]
    g0.w = (uint32_t)((ga >> 32) & 0x01FFFFFFu)        // global_addr[56:32]
         | (2u << 30);                                 // type = 2 (image)
    v8i g1;
    g1[0] = (int)(2u << 16);                           // workgroup_mask=0, data_size=2 (4B)
    g1[1] = (int)((uint32_t)(EMBED_DIM & 0xFFFF) << 16);       // tensor_dim0 lo16 @ bits[63:48]
    g1[2] = (int)((uint32_t)(VOCAB_N & 0xFFFF) << 16);         // tdim0 hi16=0 | tensor_dim1 lo16
    g1[3] = (int)(((uint32_t)VOCAB_N >> 16)                    // tensor_dim1 hi16 @ [111:96]
         | ((uint32_t)(EMBED_DIM & 0xFFFF) << 16));            // tile_dim0 @ [127:112]
    g1[4] = nrows;                                     // tile_dim1 = #valid indices; tile_dim2=0
    g1[5] = EMBED_DIM;                                 // tensor_dim0_stride lo32 (elements)
    g1[6] = 0;                                         // stride hi16 | tdim1_stride (ignored)
    g1[7] = 0;
    v4i g2 = { i0, i1, i2, i3 };                       // row_index_0..3 (32-bit index mode)
    v4i g3 = { 0, 0, 0, 0 };
    v8i g4 = { 0, 0, 0, 0, 0, 0, 0, 0 };               // unused extended group (VADDR4-class)
    __builtin_amdgcn_tensor_load_to_lds(g0, g1, g2, g3, g4, 0);
}

// ---- TDM dense store: write the nrows x 2048 LDS tile to contiguous global memory ----
__device__ __forceinline__ void tdm_store_tile(float* obase, uint32_t lds_off, int nrows) {
    uint64_t ga = (uint64_t)(uintptr_t)obase;
    v4u g0;
    g0.x = 1u;                                         // count=1, normal (dense) mode
    g0.y = lds_off;
    g0.z = (uint32_t)(ga & 0xFFFFFFFFu);
    g0.w = (uint32_t)((ga >> 32) & 0x01FFFFFFu) | (2u << 30);
    v8i g1;
    g1[0] = (int)(2u << 16);                           // data_size = 4B
    g1[1] = (int)((uint32_t)(EMBED_DIM & 0xFFFF) << 16);       // tensor_dim0 = 2048
    g1[2] = (int)((uint32_t)(nrows & 0xFFFF) << 16);           // tensor_dim1 = nrows
    g1[3] = (int)((uint32_t)(EMBED_DIM & 0xFFFF) << 16);       // tdim1 hi=0, tile_dim0 = 2048
    g1[4] = nrows;                                     // tile_dim1 = nrows; tile_dim2 = 0
    g1[5] = EMBED_DIM;                                 // tensor_dim0_stride = 2048
    g1[6] = (int)((uint32_t)(EMBED_DIM & 0xFFFF) << 16);       // tensor_dim1_stride lo16 = 2048
    g1[7] = 0;                                         // tensor_dim1_stride hi32 = 0
    v4i g2 = { 0, 0, 0, 0 };
    v4i g3 = { 0, 0, 0, 0 };
    v8i g4 = { 0, 0, 0, 0, 0, 0, 0, 0 };
    __builtin_amdgcn_tensor_store_from_lds(g0, g1, g2, g3, g4, 0);
}

// One wave32 block per ROWS_PER_BLOCK output rows. Output row r = t*N_GRAMS + g is
// contiguous, so the store side is a dense tile; the load side is a TDM row-gather
// with vocab ids as row indices (row byte offset = id * 2048 * 4 = id * 8KB).
__global__ void OEPEmbedding_49065706390109_kernel(const int* __restrict__ idx,
                                                   const float* __restrict__ weight,
                                                   float* __restrict__ out,
                                                   int num_tokens, int nrows_total) {
    __shared__ alignas(128) float tile[ROWS_PER_BLOCK * EMBED_DIM];   // 32 KB
    const uint32_t lds_off = (uint32_t)(uintptr_t)(&tile[0]);         // LDS byte offset

    const int r0 = blockIdx.x * ROWS_PER_BLOCK;
    const int nrows = min(ROWS_PER_BLOCK, nrows_total - r0);

    // Uniform (scalar) index loads: ids for output rows r0 .. r0+nrows-1.
    int id[ROWS_PER_BLOCK];
#pragma unroll
    for (int j = 0; j < ROWS_PER_BLOCK; ++j) {
        int r = r0 + j;
        if (r < nrows_total) {
            int t = r / N_GRAMS;
            int g = r - t * N_GRAMS;
            id[j] = idx[(size_t)g * num_tokens + t];
        } else {
            id[j] = 0;
        }
    }

    tdm_gather_load(weight, lds_off, nrows, id[0], id[1], id[2], id[3]);
    __builtin_amdgcn_s_wait_tensorcnt(0);              // tile resident in LDS
    tdm_store_tile(out + (size_t)r0 * EMBED_DIM, lds_off, nrows);
    // S_ENDPGM performs an implicit wait-idle, draining the outstanding tensor store.
}

extern "C" void kernel_launch(void* const* d_in, const int* in_sizes, int n_in,
                              void* d_out, int out_size, void* d_ws, size_t ws_size,
                              hipStream_t stream) {
    const int*   idx    = (const int*)d_in[0];    // [N_GRAMS, num_tokens] int32
    const float* weight = (const float*)d_in[1];  // [VOCAB, EMBED_DIM] fp32
    float*       out    = (float*)d_out;          // [num_tokens, N_GRAMS, EMBED_DIM] fp32

    const int num_tokens  = in_sizes[0] / N_GRAMS;
    const int nrows_total = num_tokens * N_GRAMS;
    const int nblocks     = (nrows_total + ROWS_PER_BLOCK - 1) / ROWS_PER_BLOCK;

    OEPEmbedding_49065706390109_kernel<<<dim3(nblocks), dim3(32), 0, stream>>>(
        idx, weight, out, num_tokens, nrows_total);
}